// WeightedPathSummationBlock_10144712753165
// MI455X (gfx1250) — compile-verified
//
#include <hip/hip_runtime.h>

// WeightedPathSummation for MI455X (gfx1250, wave32).
//
// out[n, f*3+x] = sum_p weight[species(n), p, f] * X[n, p*384 + f*3 + x]
//
// Phase 1: species gate G = Y(16x10, one-hot) @ W(10x1024) via
//          V_WMMA_F32_16X16X4_F32 (exact in f32) -> 64 KB LDS tile.
// Phase 2: bandwidth-bound stream of x with b128 loads/stores.

typedef __attribute__((ext_vector_type(2))) float v2f;
typedef __attribute__((ext_vector_type(4))) float v4f;
typedef __attribute__((ext_vector_type(8))) float v8f;

#define N_ATOMS_C   100000
#define N_SPECIES_C 10
#define N_PATHS_C   8
#define N_FEAT_C    128
#define X_ROW       4096   // full x row; we use first 3072
#define OUT_ROW     384    // N_FEAT*3
#define GCOLS       1024   // N_PATHS*N_FEAT
#define ATOMS_PER_BLOCK 16
#define THREADS     512    // 16 waves (wave32)

__global__ __launch_bounds__(THREADS)
void wps_gfx1250_kernel(const float* __restrict__ x,
                        const float* __restrict__ y,
                        const float* __restrict__ w,
                        float* __restrict__ out)
{
    // G[a][col] = weight[species(n0+a)][col], col = p*128 + f.  64 KB.
    __shared__ float G[ATOMS_PER_BLOCK * GCOLS];

    const int tid  = threadIdx.x;
    const int lane = tid & 31;
    const int wave = tid >> 5;
    const int half = lane >> 4;   // 0: lanes 0-15, 1: lanes 16-31
    const int m    = lane & 15;   // WMMA tile row (A) / column (B,C)
    const int n0   = blockIdx.x * ATOMS_PER_BLOCK;

    // ---------------- Phase 1: G = Y @ W via v_wmma_f32_16x16x4_f32 ------
    // A (16x4 f32): lanes 0-15 hold {K=k0,k0+1}, lanes 16-31 hold {K=k0+2,k0+3},
    // row M = lane%16.  K=10 padded to 12 -> 3 chained WMMAs.
    v2f A[3];
#pragma unroll
    for (int c = 0; c < 3; ++c) {
        const int kb = 4 * c + 2 * half;
        const float* yr = y + (size_t)(n0 + m) * N_SPECIES_C;
        A[c].x = (kb     < N_SPECIES_C) ? yr[kb]     : 0.0f;
        A[c].y = (kb + 1 < N_SPECIES_C) ? yr[kb + 1] : 0.0f;
    }

#pragma unroll
    for (int t = 0; t < 4; ++t) {
        const int ct  = wave * 4 + t;   // column tile 0..63
        const int col = ct * 16 + m;    // global column 0..1023
        v8f acc = {};
#pragma unroll
        for (int c = 0; c < 3; ++c) {
            const int kb = 4 * c + 2 * half;
            v2f B;  // B (4x16 f32): row K striped like A, N = lane%16
            B.x = (kb     < N_SPECIES_C) ? w[(size_t)(kb)     * GCOLS + col] : 0.0f;
            B.y = (kb + 1 < N_SPECIES_C) ? w[(size_t)(kb + 1) * GCOLS + col] : 0.0f;
            acc = __builtin_amdgcn_wmma_f32_16x16x4_f32(
                      false, A[c], false, B, (short)0, acc, false, false);
        }
        // C/D layout: VGPR v -> M=v (lanes 0-15) / M=8+v (lanes 16-31), N=lane%16
#pragma unroll
        for (int v = 0; v < 8; ++v) {
            G[(v + 8 * half) * GCOLS + col] = acc[v];
        }
    }
    __syncthreads();

    // ---------------- Phase 2: stream x, reduce over paths ---------------
    // 16 atoms * 96 float4-outputs = 1536 quads; 512 threads * 3 iters.
    // 96 quads/atom = 3 full waves -> every wave's b128 loads are one
    // contiguous 512B line; no wave straddles an atom.
#pragma unroll
    for (int iter = 0; iter < 3; ++iter) {
        const int q = iter * THREADS + tid;      // 0..1535
        const int a = q / 96;                    // atom within block tile
        const int r = q - a * 96;
        const int j = r * 4;                     // first of 4 outputs (<=380)
        const int n = n0 + a;
        const int f0 = j / 3;                    // quad spans f0 and f0+1 (<=127)
        const int thresh = 3 * (f0 + 1) - j;     // elems < thresh use f0

        const float* xp = x + (size_t)n * X_ROW + j;
        const float* gp = G + a * GCOLS + f0;

        float a0 = 0.0f, a1 = 0.0f, a2 = 0.0f, a3 = 0.0f;
#pragma unroll
        for (int p = 0; p < N_PATHS_C; ++p) {
            const v4f xv = *(const v4f*)(xp + p * OUT_ROW);  // global_load_b128
            const float g0 = gp[p * N_FEAT_C];
            const float g1 = gp[p * N_FEAT_C + 1];
            a0 += xv.x * g0;                          // elem 0 always < thresh
            a1 += xv.y * ((1 < thresh) ? g0 : g1);
            a2 += xv.z * ((2 < thresh) ? g0 : g1);
            a3 += xv.w * g1;                          // elem 3 always >= thresh
        }
        v4f o; o.x = a0; o.y = a1; o.z = a2; o.w = a3;
        *(v4f*)(out + (size_t)n * OUT_ROW + j) = o;   // global_store_b128
    }
}

extern "C" void kernel_launch(void* const* d_in, const int* in_sizes, int n_in,
                              void* d_out, int out_size, void* d_ws, size_t ws_size,
                              hipStream_t stream) {
    (void)in_sizes; (void)n_in; (void)out_size; (void)d_ws; (void)ws_size;
    const float* x = (const float*)d_in[0];   // [100000, 4096] f32
    const float* y = (const float*)d_in[1];   // [100000, 10]   f32 one-hot
    const float* w = (const float*)d_in[2];   // [10, 8, 128]   f32
    float* out     = (float*)d_out;           // [100000, 384]  f32

    const int blocks = N_ATOMS_C / ATOMS_PER_BLOCK;  // 6250, exact
    wps_gfx1250_kernel<<<blocks, THREADS, 0, stream>>>(x, y, w, out);
}